// KVEmbedding_2723009266562
// MI455X (gfx1250) — compile-verified
//
#include <hip/hip_runtime.h>
#include <hip/hip_bf16.h>
#include <stdint.h>

// ---------------------------------------------------------------------------
// KV embedding gather for MI455X (gfx1250).
//
// out[row, :] = table[indices[row], :]   row in [0, 819200), 64 fp32 per row.
//
// Pure bandwidth kernel (~426 MB traffic -> ~18us at 23.3 TB/s); no FLOPs, so
// WMMA is inapplicable. CDNA5-specific path: async global<->LDS DMA
// (ASYNCcnt-tracked GLOBAL_LOAD_ASYNC_TO_LDS_B128 /
// GLOBAL_STORE_ASYNC_FROM_LDS_B128) — data never touches VGPRs.
//
// Cache policy: unique table rows touched ~= 559K * 256B ~= 143 MB, which
// fits in the 192 MB L2. The 210 MB output stream is write-once, so the
// async store uses TH=NT (cpol=1) to avoid evicting table lines from L2;
// the table load keeps default RT so gathered rows stay resident for the
// ~1.47x average row reuse.
//
// Layout: one thread = one 16-byte chunk; 16 lanes = one 256 B row, so each
// wave32 gathers 2 rows with fully-coalesced B128 on both sides.
// ---------------------------------------------------------------------------

#define BLOCK 256
#define CHUNK_SHIFT 4          // 16 chunks (of float4) per 64-float row

#if defined(__gfx1250__) && \
    __has_builtin(__builtin_amdgcn_global_load_async_to_lds_b128) && \
    __has_builtin(__builtin_amdgcn_global_store_async_from_lds_b128)
#define USE_ASYNC_LDS 1
#else
#define USE_ASYNC_LDS 0
#endif

// gfx12-family CPol: TH in bits [2:0]; store TH=NT is 1 (CDNA5 ISA).
#define CPOL_RT       0
#define CPOL_STORE_NT 1

// 16-byte int vector type matching the builtin's pointee type.
typedef int v4i __attribute__((vector_size(16)));
typedef __attribute__((address_space(1))) v4i gbl_v4i;   // global (AS1)
typedef __attribute__((address_space(3))) v4i lds_v4i;   // LDS    (AS3)

__global__ __launch_bounds__(BLOCK) void kv_gather_kernel(
    const long long* __restrict__ indices,   // [rows] int64
    const float*     __restrict__ table,     // [VOCAB, 64] fp32
    float*           __restrict__ out,       // [rows, 64] fp32
    int nchunks)                             // rows * 16
{
#if USE_ASYNC_LDS
    __shared__ alignas(16) char staging[BLOCK * 16];
#endif

    const int c = blockIdx.x * BLOCK + threadIdx.x;
    if (c >= nchunks) return;

    const int row = c >> CHUNK_SHIFT;              // which output row
    const int sub = c & ((1 << CHUNK_SHIFT) - 1);  // which float4 within row

    const long long t = indices[row];        // 16 lanes share one 8B index load
    const float* src = table + (size_t)t * 64 + (size_t)sub * 4;
    float*       dst = out   + (size_t)c * 4;

#if USE_ASYNC_LDS
    // Stage through LDS with the CDNA5 async DMA path (ASYNCcnt-tracked).
    lds_v4i* lds_slot = (lds_v4i*)(&staging[(unsigned)threadIdx.x * 16u]);

    // global -> LDS; RT policy keeps gathered table rows resident in L2.
    __builtin_amdgcn_global_load_async_to_lds_b128(
        (gbl_v4i*)src, lds_slot, /*offset=*/0, /*cpol=*/CPOL_RT);

    // The store must not read LDS before the load lands.
#if __has_builtin(__builtin_amdgcn_s_wait_asynccnt)
    __builtin_amdgcn_s_wait_asynccnt(0);
#else
    asm volatile("s_wait_asynccnt 0" ::: "memory");
#endif

    // LDS -> global; NT policy: the output is write-once/never-read, don't
    // let it displace the ~143 MB unique table footprint from the 192 MB L2.
    // S_ENDPGM's implicit wait-idle drains the remaining ASYNCcnt.
    __builtin_amdgcn_global_store_async_from_lds_b128(
        (gbl_v4i*)dst, lds_slot, /*offset=*/0, /*cpol=*/CPOL_STORE_NT);
#else
    // Fallback: plain B128 gather through VGPRs.
    const float4 v = *(const float4*)src;
    *(float4*)dst = v;
#endif
}

extern "C" void kernel_launch(void* const* d_in, const int* in_sizes, int n_in,
                              void* d_out, int out_size, void* d_ws, size_t ws_size,
                              hipStream_t stream) {
    // setup_inputs() order: indices (int64), table (fp32), dummy (fp32[1]).
    const long long* indices = (const long long*)d_in[0];
    const float*     table   = (const float*)d_in[1];
    // dummy contributes exactly 0.0f for finite inputs -> ignored.
    float* out = (float*)d_out;

    // out_size = rows * 64 floats; one chunk = 4 floats (16 bytes).
    const int nchunks = out_size / 4;
    const int blocks  = (nchunks + BLOCK - 1) / BLOCK;

    kv_gather_kernel<<<blocks, BLOCK, 0, stream>>>(indices, table, out, nchunks);
}